// GraphEncoding_5205500363081
// MI455X (gfx1250) — compile-verified
//
#include <hip/hip_runtime.h>

#define NN   4096      // nodes
#define WIN  10        // window
#define H    96        // hidden
#define NE   131072    // edges
#define FCST 4         // forecast steps
#define WSTR 16        // window row stride (10 original + 4 preds, padded)
#define EPS  1e-5f
#define NT   3         // 16x48 C tile per wave (3 accumulators)

typedef float v2f __attribute__((ext_vector_type(2)));
typedef float v8f __attribute__((ext_vector_type(8)));

__device__ __forceinline__ float sigm(float x) { return 1.0f / (1.0f + __expf(-x)); }

// ---------------------------------------------------------------------------
// fp32 WMMA GEMM: C[M][N] = concatK(A0,A1)[M][K] @ B[K][N] (+bias)(+relu)
// A element k<K0 from A0 (row stride lda0), k>=K0 from A1 (stride lda1);
// handled as two sequential K loops (K0 uniform) so the hot loop has no
// per-iteration region selects. One wave owns a 16x48 tile: each A fragment
// (one aligned b64 load) feeds 3 v_wmma_f32_16x16x4_f32. Requirements:
// N % 48 == 0, K % 4 == 0, K0 % 4 == 0 (all satisfied by this model).
// Lane layouts per CDNA5 ISA 7.12.2:
//   A 16x4 : a[v] = A[m0 + lane15][k0 + 2*half + v]
//   B 4x16 : b[v] = B[k0 + 2*half + v][n0 + lane15]
//   C 16x16: c[v] = C[m0 + v + 8*half][n0 + lane15]
// ---------------------------------------------------------------------------
__global__ void gemm_f32_wmma(const float* __restrict__ A0, int lda0, int K0,
                              const float* __restrict__ A1, int lda1,
                              const float* __restrict__ B,
                              const float* __restrict__ bias,
                              float* __restrict__ C, int ldc,
                              int M, int N, int K, int doRelu)
{
    const int lane = threadIdx.x & 31;
    const int wave = threadIdx.x >> 5;
    const int half = lane >> 4;
    const int l15  = lane & 15;
    const int ngrp = N / (16 * NT);
    const int mtil = M >> 4;
    int g = blockIdx.x * (blockDim.x >> 5) + wave;
    if (g >= mtil * ngrp) return;                 // wave-uniform exit, EXEC stays all-1
    const int m0 = (g / ngrp) << 4;
    const int n0 = (g % ngrp) * (16 * NT);
    const int am = m0 + l15;                      // A row owned by this lane
    const int bc = n0 + l15;                      // first B/C column owned by this lane

    const v8f vzero = {0.f, 0.f, 0.f, 0.f, 0.f, 0.f, 0.f, 0.f};
    v8f acc[NT];
#pragma unroll
    for (int j = 0; j < NT; ++j) acc[j] = vzero;

    // ---- K region 1: A0 ----
    {
        const float* __restrict__ ar = A0 + (size_t)am * lda0;
        for (int k0 = 0; k0 < K0; k0 += 4) {
            const int ka = k0 + (half << 1);
            v2f a = *(const v2f*)(ar + ka);       // 8B-aligned (even stride, even ka)
            const float* __restrict__ br = B + (size_t)ka * N + bc;
#pragma unroll
            for (int j = 0; j < NT; ++j) {
                v2f b;
                b.x = br[16 * j];
                b.y = br[16 * j + N];
                acc[j] = __builtin_amdgcn_wmma_f32_16x16x4_f32(false, a, false, b,
                                                               (short)0, acc[j],
                                                               false, false);
            }
        }
    }
    // ---- K region 2: A1 (k >= K0) ----
    if (K > K0) {
        const float* __restrict__ ar = A1 + (size_t)am * lda1 - K0;
        for (int k0 = K0; k0 < K; k0 += 4) {
            const int ka = k0 + (half << 1);
            v2f a = *(const v2f*)(ar + ka);
            const float* __restrict__ br = B + (size_t)ka * N + bc;
#pragma unroll
            for (int j = 0; j < NT; ++j) {
                v2f b;
                b.x = br[16 * j];
                b.y = br[16 * j + N];
                acc[j] = __builtin_amdgcn_wmma_f32_16x16x4_f32(false, a, false, b,
                                                               (short)0, acc[j],
                                                               false, false);
            }
        }
    }

    // ---- epilogue ----
#pragma unroll
    for (int j = 0; j < NT; ++j) {
        const int col = bc + 16 * j;
        const float bs = bias ? bias[col] : 0.0f;
#pragma unroll
        for (int v = 0; v < 8; ++v) {
            float val = acc[j][v] + bs;
            if (doRelu) val = fmaxf(val, 0.0f);
            C[(size_t)(m0 + v + (half << 3)) * ldc + col] = val;
        }
    }
}

// ----------------------------- setup kernels -------------------------------
__global__ void k_deg_scatter(const long long* __restrict__ adj,
                              const float* __restrict__ ew, float* __restrict__ deg)
{
    int e = blockIdx.x * blockDim.x + threadIdx.x;
    if (e < NE) atomicAdd(&deg[(int)adj[NE + e]], ew[e]);
}

__global__ void k_selfnorm(const float* __restrict__ deg,
                           float* __restrict__ dinv, float* __restrict__ self_norm)
{
    int n = blockIdx.x * blockDim.x + threadIdx.x;
    if (n < NN) { float d = rsqrtf(deg[n] + 1.0f); dinv[n] = d; self_norm[n] = d * d; }
}

__global__ void k_edgenorm(const long long* __restrict__ adj, const float* __restrict__ ew,
                           const float* __restrict__ dinv, float* __restrict__ norm)
{
    int e = blockIdx.x * blockDim.x + threadIdx.x;
    if (e < NE) norm[e] = dinv[(int)adj[e]] * ew[e] * dinv[(int)adj[NE + e]];
}

__global__ void k_initwin(const float* __restrict__ x, float* __restrict__ win)
{
    int tid = blockIdx.x * blockDim.x + threadIdx.x;
    if (tid < NN * WIN) win[(tid / WIN) * WSTR + (tid % WIN)] = x[tid];
}

__global__ void k_bnprep(const float* g1, const float* b1, const float* m1, const float* v1,
                         const float* g2, const float* b2, const float* m2, const float* v2,
                         float* __restrict__ bn)   // [s1|sh1|s2|sh2] x 96
{
    int f = threadIdx.x;
    if (f < H) {
        float s1 = g1[f] * rsqrtf(v1[f] + EPS);
        bn[f] = s1;           bn[H + f]     = b1[f] - m1[f] * s1;
        float s2 = g2[f] * rsqrtf(v2[f] + EPS);
        bn[2 * H + f] = s2;   bn[3 * H + f] = b2[f] - m2[f] * s2;
    }
}

// combined recurrent weight, layer1: Wc[288][384] = [wih.T ; whh.T]
__global__ void k_build_wr1(const float* __restrict__ wih, const float* __restrict__ whh,
                            float* __restrict__ Wc)
{
    int tid = blockIdx.x * blockDim.x + threadIdx.x;
    if (tid < 288 * 384) {
        int g = tid % 384, k = tid / 384;
        Wc[tid] = (k < 192) ? wih[(size_t)g * 192 + k] : whh[(size_t)g * 96 + (k - 192)];
    }
}
// layer2: Wc[192][384]
__global__ void k_build_wr2(const float* __restrict__ wih, const float* __restrict__ whh,
                            float* __restrict__ Wc)
{
    int tid = blockIdx.x * blockDim.x + threadIdx.x;
    if (tid < 192 * 384) {
        int g = tid % 384, k = tid / 384;
        Wc[tid] = (k < 96) ? wih[(size_t)g * 96 + k] : whh[(size_t)g * 96 + (k - 96)];
    }
}
// fc1 weight transposed + zero-padded K: Wc[208][96]
__global__ void k_build_wfc(const float* __restrict__ w, float* __restrict__ Wc)
{
    int tid = blockIdx.x * blockDim.x + threadIdx.x;
    if (tid < 208 * 96) {
        int f = tid % 96, k = tid / 96;
        Wc[tid] = (k < 202) ? w[(size_t)f * 202 + k] : 0.0f;
    }
}
__global__ void k_bias_sum(const float* a, const float* b, float* o, int n)
{
    int i = blockIdx.x * blockDim.x + threadIdx.x;
    if (i < n) o[i] = a[i] + b[i];
}

// ----------------------------- per-step kernels ----------------------------
// GCN1 is rank-1 in features: scatter scalars only.
__global__ void k_scatter1(const long long* __restrict__ adj, const float* __restrict__ norm,
                           const float* __restrict__ win, float* __restrict__ aggscal, int s)
{
    int tid = blockIdx.x * blockDim.x + threadIdx.x;
    if (tid < NE * WIN) {
        int e = tid & (NE - 1), t = tid >> 17;
        int src = (int)adj[e], dst = (int)adj[NE + e];
        atomicAdd(&aggscal[t * NN + dst], win[src * WSTR + s + t] * norm[e]);
    }
}

// h1 = bn1(relu((aggscal + x*self_norm)*W1[f] + b1[f])) -> hcat[..][0:96]
__global__ void k_gcn1(const float* __restrict__ aggscal, const float* __restrict__ win,
                       const float* __restrict__ self_norm, const float* __restrict__ W1,
                       const float* __restrict__ b1, const float* __restrict__ bn,
                       float* __restrict__ hcat, int s)
{
    int tid = blockIdx.x * blockDim.x + threadIdx.x;
    if (tid < WIN * NN * H) {
        int f = tid % H, r = tid / H;
        int node = r & (NN - 1), t = r >> 12;
        float v = aggscal[t * NN + node] + win[node * WSTR + s + t] * self_norm[node];
        float h = fmaxf(v * W1[f] + b1[f], 0.0f);
        hcat[(size_t)r * 192 + f] = h * bn[f] + bn[H + f];
    }
}

// full 96-wide edge scatter for GCN2
__global__ void k_scatter2(const long long* __restrict__ adj, const float* __restrict__ norm,
                           const float* __restrict__ hw2, float* __restrict__ agg2)
{
    int tid = blockIdx.x * blockDim.x + threadIdx.x;
    if (tid < NE * WIN * H) {
        int f = tid % H, r = tid / H;
        int e = r & (NE - 1), t = r >> 17;
        int src = (int)adj[e], dst = (int)adj[NE + e];
        atomicAdd(&agg2[((size_t)t * NN + dst) * H + f],
                  hw2[((size_t)t * NN + src) * H + f] * norm[e]);
    }
}

// h2 = bn2(relu(agg2 + hw2*self_norm + b2)) -> hcat[..][96:192]
__global__ void k_gcn2(const float* __restrict__ agg2, const float* __restrict__ hw2,
                       const float* __restrict__ self_norm, const float* __restrict__ b2,
                       const float* __restrict__ bn, float* __restrict__ hcat)
{
    int tid = blockIdx.x * blockDim.x + threadIdx.x;
    if (tid < WIN * NN * H) {
        int f = tid % H, r = tid / H;
        int node = r & (NN - 1);
        float v = agg2[tid] + hw2[tid] * self_norm[node] + b2[f];
        float h = fmaxf(v, 0.0f);
        hcat[(size_t)r * 192 + H + f] = h * bn[2 * H + f] + bn[3 * H + f];
    }
}

// LSTM cell: gates [i|f|g|o] per node (ld 384); h,c updated in place
__global__ void k_lstm_cell(const float* __restrict__ gates,
                            float* __restrict__ h, float* __restrict__ c,
                            float* __restrict__ y)
{
    int tid = blockIdx.x * blockDim.x + threadIdx.x;
    if (tid < NN * H) {
        int f = tid % H, node = tid / H;
        const float* g = gates + (size_t)node * 384;
        float ig = sigm(g[f]);
        float fg = sigm(g[H + f]);
        float gg = tanhf(g[2 * H + f]);
        float og = sigm(g[3 * H + f]);
        float cc = fg * c[tid] + ig * gg;
        float hh = og * tanhf(cc);
        c[tid] = cc; h[tid] = hh;
        if (y) y[tid] = hh;
    }
}

// zcat[node][208] = [y1_last(96) | y2_last(96) | skip(10) | pad(6)=0]
__global__ void k_zcat(const float* __restrict__ y1last, const float* __restrict__ h2,
                       const float* __restrict__ win, float* __restrict__ zcat, int s)
{
    int tid = blockIdx.x * blockDim.x + threadIdx.x;
    if (tid < NN * 208) {
        int k = tid % 208, node = tid / 208;
        float v;
        if (k < 96)       v = y1last[node * H + k];
        else if (k < 192) v = h2[node * H + (k - 96)];
        else if (k < 202) v = win[node * WSTR + s + (k - 192)];
        else              v = 0.0f;
        zcat[tid] = v;
    }
}

// fc2 (96->1) + relu, update window and write output
__global__ void k_fc2(const float* __restrict__ fc1out, const float* __restrict__ w,
                      const float* __restrict__ b, float* __restrict__ win,
                      float* __restrict__ out, int s)
{
    int node = blockIdx.x * blockDim.x + threadIdx.x;
    if (node < NN) {
        float acc = b[0];
        const float* row = fc1out + (size_t)node * H;
#pragma unroll 8
        for (int f = 0; f < H; ++f) acc += row[f] * w[f];
        acc = fmaxf(acc, 0.0f);
        win[node * WSTR + WIN + s] = acc;
        out[node * FCST + s] = acc;
    }
}

// ---------------------------------------------------------------------------
static inline int cdiv(int a, int b) { return (a + b - 1) / b; }

extern "C" void kernel_launch(void* const* d_in, const int* in_sizes, int n_in,
                              void* d_out, int out_size, void* d_ws, size_t ws_size,
                              hipStream_t stream)
{
    (void)in_sizes; (void)n_in; (void)out_size; (void)ws_size;
    const float*     x     = (const float*)d_in[0];
    const long long* adj   = (const long long*)d_in[1];
    const float*     ew    = (const float*)d_in[2];
    const float*     W1    = (const float*)d_in[3];
    const float*     b1    = (const float*)d_in[4];
    const float*     W2    = (const float*)d_in[5];
    const float*     b2    = (const float*)d_in[6];
    const float*     fc1_b = (const float*)d_in[24];
    const float*     fc2_w = (const float*)d_in[25];
    const float*     fc2_b = (const float*)d_in[26];
    float* out = (float*)d_out;

    // ---- workspace layout (floats, 256B-aligned chunks); total ~95 MB ----
    float* W = (float*)d_ws;
    size_t off = 0;
    auto alloc = [&](size_t n) { float* p = W + off; off += (n + 63) & ~(size_t)63; return p; };
    float* deg      = alloc(NN);
    float* dinv     = alloc(NN);
    float* selfn    = alloc(NN);
    float* norm     = alloc(NE);
    float* win      = alloc((size_t)NN * WSTR);
    float* aggscal  = alloc((size_t)WIN * NN);
    float* hcat     = alloc((size_t)WIN * NN * 192);   // [h1|h2] per (t,node)
    float* hw2      = alloc((size_t)WIN * NN * H);
    float* agg2     = alloc((size_t)WIN * NN * H);
    float* y1       = alloc((size_t)WIN * NN * H);
    float* hbuf     = alloc((size_t)NN * H);
    float* cbuf     = alloc((size_t)NN * H);
    float* gates    = alloc((size_t)NN * 384);
    float* zcat     = alloc((size_t)NN * 208);
    float* fc1out   = alloc((size_t)NN * H);
    float* Wr1c     = alloc(288 * 384);
    float* Wr2c     = alloc(192 * 384);
    float* Wfc      = alloc(208 * 96);
    float* rbias1   = alloc(384);
    float* rbias2   = alloc(384);
    float* bnp      = alloc(4 * H);

    auto gemm = [&](const float* A0, int lda0, int K0, const float* A1, int lda1,
                    const float* B, const float* bias, float* C, int ldc,
                    int M, int N, int K, int relu) {
        int waves = (M / 16) * (N / (16 * NT));
        gemm_f32_wmma<<<cdiv(waves, 8), 256, 0, stream>>>(
            A0, lda0, K0, A1 ? A1 : A0, A1 ? lda1 : lda0,
            B, bias, C, ldc, M, N, K, relu);
    };

    // ---- one-time setup (per call; deterministic) ----
    hipMemsetAsync(deg, 0, NN * sizeof(float), stream);
    k_deg_scatter<<<cdiv(NE, 256), 256, 0, stream>>>(adj, ew, deg);
    k_selfnorm<<<cdiv(NN, 256), 256, 0, stream>>>(deg, dinv, selfn);
    k_edgenorm<<<cdiv(NE, 256), 256, 0, stream>>>(adj, ew, dinv, norm);
    k_initwin<<<cdiv(NN * WIN, 256), 256, 0, stream>>>(x, win);
    k_bnprep<<<1, 128, 0, stream>>>((const float*)d_in[7],  (const float*)d_in[8],
                                    (const float*)d_in[9],  (const float*)d_in[10],
                                    (const float*)d_in[11], (const float*)d_in[12],
                                    (const float*)d_in[13], (const float*)d_in[14], bnp);
    k_build_wr1<<<cdiv(288 * 384, 256), 256, 0, stream>>>((const float*)d_in[15],
                                                          (const float*)d_in[16], Wr1c);
    k_build_wr2<<<cdiv(192 * 384, 256), 256, 0, stream>>>((const float*)d_in[19],
                                                          (const float*)d_in[20], Wr2c);
    k_build_wfc<<<cdiv(208 * 96, 256), 256, 0, stream>>>((const float*)d_in[23], Wfc);
    k_bias_sum<<<2, 256, 0, stream>>>((const float*)d_in[17], (const float*)d_in[18], rbias1, 384);
    k_bias_sum<<<2, 256, 0, stream>>>((const float*)d_in[21], (const float*)d_in[22], rbias2, 384);

    // ---- forecast loop ----
    for (int s = 0; s < FCST; ++s) {
        // GCN1 (rank-1 scatter) -> hcat[:, :, 0:96]
        hipMemsetAsync(aggscal, 0, (size_t)WIN * NN * sizeof(float), stream);
        k_scatter1<<<cdiv(NE * WIN, 256), 256, 0, stream>>>(adj, norm, win, aggscal, s);
        k_gcn1<<<cdiv(WIN * NN * H, 256), 256, 0, stream>>>(aggscal, win, selfn, W1, b1,
                                                            bnp, hcat, s);
        // GCN2: hw2 = h1 @ W2 (WMMA), edge scatter, combine -> hcat[:, :, 96:192]
        gemm(hcat, 192, H, nullptr, 0, W2, nullptr, hw2, H, WIN * NN, H, H, 0);
        hipMemsetAsync(agg2, 0, (size_t)WIN * NN * H * sizeof(float), stream);
        k_scatter2<<<cdiv(NE * WIN * H, 256), 256, 0, stream>>>(adj, norm, hw2, agg2);
        k_gcn2<<<cdiv(WIN * NN * H, 256), 256, 0, stream>>>(agg2, hw2, selfn, b2, bnp, hcat);

        // LSTM layer 1: gates = [hcat_t | h] @ Wr1c + (bih+bhh)  (K = 192+96)
        hipMemsetAsync(hbuf, 0, (size_t)NN * H * sizeof(float), stream);
        hipMemsetAsync(cbuf, 0, (size_t)NN * H * sizeof(float), stream);
        for (int t = 0; t < WIN; ++t) {
            gemm(hcat + (size_t)t * NN * 192, 192, 192, hbuf, H,
                 Wr1c, rbias1, gates, 384, NN, 384, 288, 0);
            k_lstm_cell<<<cdiv(NN * H, 256), 256, 0, stream>>>(gates, hbuf, cbuf,
                                                               y1 + (size_t)t * NN * H);
        }
        // LSTM layer 2: gates = [y1_t | h] @ Wr2c + (bih+bhh)  (K = 96+96)
        hipMemsetAsync(hbuf, 0, (size_t)NN * H * sizeof(float), stream);
        hipMemsetAsync(cbuf, 0, (size_t)NN * H * sizeof(float), stream);
        for (int t = 0; t < WIN; ++t) {
            gemm(y1 + (size_t)t * NN * H, H, H, hbuf, H,
                 Wr2c, rbias2, gates, 384, NN, 384, 192, 0);
            k_lstm_cell<<<cdiv(NN * H, 256), 256, 0, stream>>>(gates, hbuf, cbuf, nullptr);
        }
        // head: zcat -> fc1 (WMMA, relu) -> fc2 -> pred, window update
        k_zcat<<<cdiv(NN * 208, 256), 256, 0, stream>>>(y1 + (size_t)9 * NN * H, hbuf,
                                                        win, zcat, s);
        gemm(zcat, 208, 208, nullptr, 0, Wfc, fc1_b, fc1out, H, NN, H, 208, 1);
        k_fc2<<<cdiv(NN, 256), 256, 0, stream>>>(fc1out, fc2_w, fc2_b, win, out, s);
    }
}